// WaveletLayer_70497593197413
// MI455X (gfx1250) — compile-verified
//
#include <hip/hip_runtime.h>
#include <hip/hip_bf16.h>

#define THREADS 256
constexpr int ROW_N = 4096;   // row length
constexpr int COEF_L = 2049;  // number of wavelet coefficients per row

#if defined(__AMDGCN__) && __has_builtin(__builtin_amdgcn_global_load_async_to_lds_b128)
#define USE_ASYNC_LDS 1
#else
#define USE_ASYNC_LDS 0
#endif

#if USE_ASYNC_LDS
typedef int vint4 __attribute__((__vector_size__(4 * sizeof(int))));
typedef __attribute__((address_space(1))) vint4* gas1_v4i;
typedef __attribute__((address_space(3))) vint4* gas3_v4i;
#endif

// symmetric boundary reflection for x index in [-2, ROW_N+1]
__device__ __forceinline__ int refl4096(int i) {
    i = (i < 0) ? (-1 - i) : i;                    // x[-1]->x[0], x[-2]->x[1]
    i = (i > ROW_N - 1) ? (2 * ROW_N - 1 - i) : i; // x[4096]->x[4095], x[4097]->x[4094]
    return i;
}

__global__ __launch_bounds__(THREADS) void wavelet_fused_kernel(
    const float* __restrict__ x, const float* __restrict__ kern, float* __restrict__ y)
{
    __shared__ __align__(16) float sx[ROW_N];       // staged x row
    __shared__ __align__(16) float sa[COEF_L + 3];  // modulated cA
    __shared__ __align__(16) float sd[COEF_L + 3];  // modulated cD

    const int row = blockIdx.x;
    const int tid = threadIdx.x;
    const float* __restrict__ xrow = x + (size_t)row * ROW_N;
    const float* __restrict__ krow = kern + (size_t)row * COEF_L;
    float* __restrict__ yrow = y + (size_t)row * ROW_N;

#if defined(__AMDGCN__)
    // warm the kernel row into cache while we stage x (global_prefetch_b8)
    if (tid * 32 < COEF_L) __builtin_prefetch(krow + tid * 32, 0, 1);
#endif

    // ---------------- Phase 1: stage x row into LDS (16B per lane) -------------
#if USE_ASYNC_LDS
#pragma unroll
    for (int v = 0; v < ROW_N / (THREADS * 4); ++v) {
        const int q = v * THREADS + tid;  // float4 index
        __builtin_amdgcn_global_load_async_to_lds_b128(
            (gas1_v4i)(xrow + q * 4),
            (gas3_v4i)(&sx[q * 4]),
            0, 0);
    }
#if __has_builtin(__builtin_amdgcn_s_wait_asynccnt)
    __builtin_amdgcn_s_wait_asynccnt(0);
#else
    asm volatile("s_wait_asynccnt 0" ::: "memory");
#endif
#else
#pragma unroll
    for (int v = 0; v < ROW_N / (THREADS * 4); ++v) {
        const int q = v * THREADS + tid;
        ((float4*)sx)[q] = ((const float4*)xrow)[q];
    }
#endif
    __syncthreads();

    // db2 decomposition filters (dec_lo / dec_hi from the reference)
    constexpr float lo0 = -0.12940952255126037f, lo1 = 0.22414386804185735f,
                    lo2 =  0.8365163037378079f,  lo3 = 0.48296291314453416f;
    constexpr float hi0 = -0.48296291314453416f, hi1 = 0.8365163037378079f,
                    hi2 = -0.22414386804185735f, hi3 = -0.12940952255126037f;

    // ---------------- Phase 2: DWT + per-position modulation -------------------
    // cA[t] = lo3*x[2t-2] + lo2*x[2t-1] + lo1*x[2t] + lo0*x[2t+1]  (reflected)
#pragma unroll
    for (int it = 0; it < (COEF_L + THREADS - 1) / THREADS; ++it) {
        const int t = it * THREADS + tid;
        if (t < COEF_L) {
            const int i = 2 * t;
            const float xm2 = sx[refl4096(i - 2)];
            const float xm1 = sx[refl4096(i - 1)];
            const float x0  = sx[refl4096(i)];
            const float x1  = sx[refl4096(i + 1)];
            const float k   = krow[t];
            const float a = lo3 * xm2 + lo2 * xm1 + lo1 * x0 + lo0 * x1;
            const float d = hi3 * xm2 + hi2 * xm1 + hi1 * x0 + hi0 * x1;
            sa[t] = a * k;
            sd[t] = d * k;
        }
    }
    __syncthreads();

    // ---------------- Phase 3: inverse DWT + ReLU, float4 stores ---------------
    // even m: lo1*a[h] + lo3*a[h+1] + hi1*d[h] + hi3*d[h+1]  (h = m/2)
    // odd  m: lo0*a[h] + lo2*a[h+1] + hi0*d[h] + hi2*d[h+1]  (h = (m-1)/2)
#pragma unroll
    for (int c = 0; c < ROW_N / (THREADS * 4); ++c) {
        const int q = c * THREADS + tid;   // float4 index; m0 = 4q (even)
        const int h = 2 * q;
        const float a0 = sa[h], a1 = sa[h + 1], a2 = sa[h + 2];
        const float d0 = sd[h], d1 = sd[h + 1], d2 = sd[h + 2];
        float4 o;
        o.x = lo1 * a0 + lo3 * a1 + hi1 * d0 + hi3 * d1;
        o.y = lo0 * a0 + lo2 * a1 + hi0 * d0 + hi2 * d1;
        o.z = lo1 * a1 + lo3 * a2 + hi1 * d1 + hi3 * d2;
        o.w = lo0 * a1 + lo2 * a2 + hi0 * d1 + hi2 * d2;
        o.x = fmaxf(o.x, 0.0f);
        o.y = fmaxf(o.y, 0.0f);
        o.z = fmaxf(o.z, 0.0f);
        o.w = fmaxf(o.w, 0.0f);
        ((float4*)yrow)[q] = o;
    }
}

extern "C" void kernel_launch(void* const* d_in, const int* in_sizes, int n_in,
                              void* d_out, int out_size, void* d_ws, size_t ws_size,
                              hipStream_t stream) {
    (void)n_in; (void)out_size; (void)d_ws; (void)ws_size;
    const float* x = (const float*)d_in[0];
    const float* k = (const float*)d_in[1];
    float* y = (float*)d_out;
    const int B = in_sizes[0] / ROW_N;  // 4096 rows
    wavelet_fused_kernel<<<dim3(B), dim3(THREADS), 0, stream>>>(x, k, y);
}